// Quanvolution__gen413_65481071397702
// MI455X (gfx1250) — compile-verified
//
#include <hip/hip_runtime.h>
#include <hip/hip_bf16.h>

// ---------------------------------------------------------------------------
// Quanvolution on MI455X (gfx1250).
//
// Math note: the 4-qubit circuit (RY product state -> CNOT chain -> RZ ->
// CNOT chain -> RZ -> <Z_i>) contains only permutations and diagonal phases
// applied to an initially-real product state, so probabilities are
// phase-independent; the per-patch features reduce EXACTLY to:
//   [cos x0, cos x1, cos x0*cos x2, cos x1*cos x3]
// (the params input is mathematically irrelevant to the output).
// ---------------------------------------------------------------------------

typedef __attribute__((ext_vector_type(2))) float v2f;
typedef __attribute__((ext_vector_type(8))) float v8f;
typedef __attribute__((ext_vector_type(4))) int v4i;

typedef __attribute__((address_space(1))) v4i gv4i;  // global 16B block
typedef __attribute__((address_space(3))) v4i lv4i;  // LDS 16B block

// 16-byte global->LDS copy: async (ASYNCcnt-tracked, VGPR-bypassing) when the
// gfx1250 builtin is available, synchronous fallback otherwise.
__device__ __forceinline__ void copy16_g2l(void* lds_dst, const void* gsrc) {
#if __has_builtin(__builtin_amdgcn_global_load_async_to_lds_b128)
  __builtin_amdgcn_global_load_async_to_lds_b128(
      (gv4i*)gsrc, (lv4i*)lds_dst, 0, 0);
#else
  *(float4*)lds_dst = *(const float4*)gsrc;
#endif
}

__device__ __forceinline__ void wait_async0() {
#if __has_builtin(__builtin_amdgcn_s_wait_asynccnt)
  __builtin_amdgcn_s_wait_asynccnt(0);
#else
  asm volatile("s_wait_asynccnt 0x0" ::: "memory");
#endif
}

// ---------------------------------------------------------------------------
// Kernel 1: patch features. 8192*196 patches, one thread per patch.
// ---------------------------------------------------------------------------
__global__ __launch_bounds__(256) void feat_kernel(const float* __restrict__ x,
                                                   float* __restrict__ feats,
                                                   int total) {
  int gp = blockIdx.x * 256 + threadIdx.x;
  if (gp >= total) return;
  int b  = gp / 196;
  int pi = gp - b * 196;
  int p  = pi / 14;
  int q  = pi - p * 14;
  const float* base = x + b * 784 + p * 56 + q * 2;  // row 2p, col 2q
  float2 r0 = *(const float2*)(base);
  float2 r1 = *(const float2*)(base + 28);
  float c0 = __cosf(r0.x);
  float c1 = __cosf(r0.y);
  float c2 = __cosf(r1.x);
  float c3 = __cosf(r1.y);
  float4 f;
  f.x = c0;
  f.y = c1;
  f.z = c0 * c2;
  f.w = c1 * c3;
  *(float4*)(feats + gp * 4) = f;
}

// ---------------------------------------------------------------------------
// Kernel 2/3: WMMA f32 GEMM  C = act(A[M x K] @ W[N x K]^T + bias).
// Block: 256 threads = 8 waves. Tile: M=64 x N=NTOT (full N), K chunked by 16.
// Both A and W chunks are staged to LDS row-major ([row][k], row padded to 20
// floats: 16B-aligned rows + conflict-free 64-bank access), double-buffered
// with async global->LDS copies pipelined against the WMMA loop.
// Wave w: rows (w>>2)*32..+32 (2 m-subtiles), cols (w&3)*(NTOT/4)
// (NTOT/64 n-subtiles).
// ---------------------------------------------------------------------------
template <int NTOT, int KTOT, bool RELU>
__global__ __launch_bounds__(256) void gemm_bias_act(
    const float* __restrict__ A, const float* __restrict__ W,
    const float* __restrict__ bias, float* __restrict__ C) {
  constexpr int NT   = NTOT / 64;  // n-tiles per wave
  constexpr int PAD  = 20;         // floats per padded 16-float row
  __shared__ float lA[2][64 * PAD];
  __shared__ float lB[2][NTOT * PAD];

  const int t     = threadIdx.x;
  const int lane  = t & 31;
  const int wv    = t >> 5;
  const int mwave = wv >> 2;          // 0..1
  const int nwave = wv & 3;           // 0..3
  const int l16   = lane & 15;
  const int kk    = (lane >> 4) * 2;  // 0 or 2 (f32 16x16x4 A/B frag layout)
  const int mbase = blockIdx.x * 64;

  v8f acc[2][NT];
#pragma unroll
  for (int i = 0; i < 2; ++i)
#pragma unroll
    for (int j = 0; j < NT; ++j) acc[i][j] = (v8f){0, 0, 0, 0, 0, 0, 0, 0};

  // Stage one K=16 chunk (A: 64x16, W: NTOTx16) into buffer `buf`.
  const int arow  = t >> 2;
  const int apart = (t & 3) * 4;
  auto stage = [&](int buf, int k0) {
    copy16_g2l(&lA[buf][arow * PAD + apart],
               A + (mbase + arow) * KTOT + k0 + apart);
#pragma unroll
    for (int i = t; i < NTOT * 4; i += 256) {
      int n = i >> 2;
      int part = (i & 3) * 4;
      copy16_g2l(&lB[buf][n * PAD + part], W + n * KTOT + k0 + part);
    }
  };

  stage(0, 0);  // prologue

  constexpr int NC = KTOT / 16;
  for (int c = 0; c < NC; ++c) {
    const int ping = c & 1;
    wait_async0();
    __syncthreads();  // buffer `ping` fully resident
    if (c + 1 < NC) stage(1 - ping, (c + 1) * 16);  // overlap next fill

#pragma unroll
    for (int ks = 0; ks < 4; ++ks) {
      const int kl = ks * 4;
      v2f afrag[2];
#pragma unroll
      for (int mt = 0; mt < 2; ++mt) {
        const float* ap =
            &lA[ping][(mwave * 32 + mt * 16 + l16) * PAD + kl + kk];
        float2 av = *(const float2*)ap;
        afrag[mt].x = av.x;
        afrag[mt].y = av.y;
      }
      v2f bfrag[NT];
#pragma unroll
      for (int nt = 0; nt < NT; ++nt) {
        const int col = nwave * (NTOT / 4) + nt * 16 + l16;
        const float* bp = &lB[ping][col * PAD + kl + kk];
        float2 bv = *(const float2*)bp;
        bfrag[nt].x = bv.x;
        bfrag[nt].y = bv.y;
      }
#pragma unroll
      for (int mt = 0; mt < 2; ++mt)
#pragma unroll
        for (int nt = 0; nt < NT; ++nt)
          acc[mt][nt] = __builtin_amdgcn_wmma_f32_16x16x4_f32(
              false, afrag[mt], false, bfrag[nt], (short)0, acc[mt][nt],
              false, false);
    }
    __syncthreads();  // all reads of `ping` done before it is refilled
  }

  // --- epilogue: bias + activation + store ---
  const int rhalf = (lane >> 4) * 8;  // C/D layout: VGPR r -> rows r, r+8
#pragma unroll
  for (int mt = 0; mt < 2; ++mt) {
#pragma unroll
    for (int nt = 0; nt < NT; ++nt) {
      const int col = nwave * (NTOT / 4) + nt * 16 + l16;
      const float bv = bias[col];
#pragma unroll
      for (int r = 0; r < 8; ++r) {
        int row = mbase + mwave * 32 + mt * 16 + r + rhalf;
        float v = acc[mt][nt][r] + bv;
        if (RELU) v = fmaxf(v, 0.0f);
        C[row * NTOT + col] = v;
      }
    }
  }
}

// ---------------------------------------------------------------------------
// Kernel 4: final 128->10 GEMV + log_softmax. One wave per row.
// ---------------------------------------------------------------------------
__global__ __launch_bounds__(256) void head_ls(const float* __restrict__ H,
                                               const float* __restrict__ W3,
                                               const float* __restrict__ b3,
                                               float* __restrict__ out,
                                               int M) {
  int wv = threadIdx.x >> 5;
  int lane = threadIdx.x & 31;
  int row = blockIdx.x * 8 + wv;
  if (row >= M) return;

  float4 h = *(const float4*)(H + row * 128 + lane * 4);

  float lg[10];
#pragma unroll
  for (int n = 0; n < 10; ++n) {
    float4 w = *(const float4*)(W3 + n * 128 + lane * 4);
    float p = h.x * w.x + h.y * w.y + h.z * w.z + h.w * w.w;
    p += __shfl_xor(p, 16);
    p += __shfl_xor(p, 8);
    p += __shfl_xor(p, 4);
    p += __shfl_xor(p, 2);
    p += __shfl_xor(p, 1);
    lg[n] = p + b3[n];
  }
  float mx = lg[0];
#pragma unroll
  for (int n = 1; n < 10; ++n) mx = fmaxf(mx, lg[n]);
  float s = 0.0f;
#pragma unroll
  for (int n = 0; n < 10; ++n) s += __expf(lg[n] - mx);
  float lse = mx + __logf(s);

  if (lane < 10) {
    float myv = 0.0f;
#pragma unroll
    for (int n = 0; n < 10; ++n)
      if (lane == n) myv = lg[n];
    out[row * 10 + lane] = myv - lse;
  }
}

// ---------------------------------------------------------------------------
extern "C" void kernel_launch(void* const* d_in, const int* in_sizes, int n_in,
                              void* d_out, int out_size, void* d_ws,
                              size_t ws_size, hipStream_t stream) {
  const float* x  = (const float*)d_in[0];  // (8192,1,28,28)
  // d_in[1] = params (2,4): provably irrelevant to the output (see header).
  const float* W1 = (const float*)d_in[2];  // (256,784)
  const float* b1 = (const float*)d_in[3];
  const float* W2 = (const float*)d_in[4];  // (128,256)
  const float* b2 = (const float*)d_in[5];
  const float* W3 = (const float*)d_in[6];  // (10,128)
  const float* b3 = (const float*)d_in[7];
  float* out = (float*)d_out;

  const int B = 8192;
  float* feats = (float*)d_ws;            // B x 784
  float* h1 = feats + (size_t)B * 784;    // B x 256
  float* h2 = h1 + (size_t)B * 256;       // B x 128

  const int total_patches = B * 196;  // 1,605,632 = 6272 * 256
  feat_kernel<<<total_patches / 256, 256, 0, stream>>>(x, feats, total_patches);

  gemm_bias_act<256, 784, true><<<B / 64, 256, 0, stream>>>(feats, W1, b1, h1);
  gemm_bias_act<128, 256, true><<<B / 64, 256, 0, stream>>>(h1, W2, b2, h2);

  head_ls<<<B / 8, 256, 0, stream>>>(h2, W3, b3, out, B);
}